// ObjectMatchingModule_52544629900002
// MI455X (gfx1250) — compile-verified
//
#include <hip/hip_runtime.h>

typedef __attribute__((ext_vector_type(2))) float v2f;
typedef __attribute__((ext_vector_type(8))) float v8f;

#define CDIM 512
#define MDIM 2048
#define NAUG 2049
#define EPS_INV 10.0f   // 1 / SINKHORN_EPS

// ---------------------------------------------------------------------------
// K0 = exp( (Q * R^T) / eps ), written into the P_aug region (ld = 2049).
// One wave computes a 16x64 tile via V_WMMA_F32_16X16X4_F32 (fp32 WMMA).
// Block = 256 threads (8 waves) covers 128 rows x 64 cols. Grid (32,16).
// A 16x4 f32 layout: lane<16 -> K={0,1}, lane>=16 -> K={2,3}; M = lane%16.
// B 4x16 f32 layout mirrors A with N = lane%16.
// C/D 16x16 f32: VGPR d, lane l -> M = d + 8*(l/16), N = l%16.
// ---------------------------------------------------------------------------
__global__ __launch_bounds__(256) void sinkhorn_gemm_exp(
    const float* __restrict__ Q, const float* __restrict__ R,
    float* __restrict__ K0)
{
    const int lane = threadIdx.x & 31;
    const int wave = threadIdx.x >> 5;
    const int l16  = lane & 15;
    const int hi   = lane >> 4;            // K-pair select (A/B); +8 rows (C/D)
    const int m0   = blockIdx.y * 128 + wave * 16;
    const int n0   = blockIdx.x * 64;

    const float* qrow  = Q + (size_t)(m0 + l16) * CDIM + 2 * hi;
    const float* rrow0 = R + (size_t)(n0 +  0 + l16) * CDIM + 2 * hi;
    const float* rrow1 = R + (size_t)(n0 + 16 + l16) * CDIM + 2 * hi;
    const float* rrow2 = R + (size_t)(n0 + 32 + l16) * CDIM + 2 * hi;
    const float* rrow3 = R + (size_t)(n0 + 48 + l16) * CDIM + 2 * hi;

    v8f acc0 = {}, acc1 = {}, acc2 = {}, acc3 = {};

    for (int k = 0; k < CDIM; k += 4) {
        v2f a  = *(const v2f*)(qrow  + k);
        v2f b0 = *(const v2f*)(rrow0 + k);
        v2f b1 = *(const v2f*)(rrow1 + k);
        v2f b2 = *(const v2f*)(rrow2 + k);
        v2f b3 = *(const v2f*)(rrow3 + k);
        acc0 = __builtin_amdgcn_wmma_f32_16x16x4_f32(false, a, false, b0, (short)0, acc0, false, false);
        acc1 = __builtin_amdgcn_wmma_f32_16x16x4_f32(false, a, false, b1, (short)0, acc1, false, false);
        acc2 = __builtin_amdgcn_wmma_f32_16x16x4_f32(false, a, false, b2, (short)0, acc2, false, false);
        acc3 = __builtin_amdgcn_wmma_f32_16x16x4_f32(false, a, false, b3, (short)0, acc3, false, false);
    }

    const int mrow = m0 + hi * 8;
    #pragma unroll
    for (int d = 0; d < 8; ++d) {
        K0[(size_t)(mrow + d) * NAUG + (n0 +  0 + l16)] = __expf(acc0[d] * EPS_INV);
        K0[(size_t)(mrow + d) * NAUG + (n0 + 16 + l16)] = __expf(acc1[d] * EPS_INV);
        K0[(size_t)(mrow + d) * NAUG + (n0 + 32 + l16)] = __expf(acc2[d] * EPS_INV);
        K0[(size_t)(mrow + d) * NAUG + (n0 + 48 + l16)] = __expf(acc3[d] * EPS_INV);
    }
}

// Fill augmented border (last row + last col) with exp(z/eps); v = 1.
__global__ void sinkhorn_init(float* __restrict__ K0, float* __restrict__ v,
                              const float* __restrict__ zp)
{
    int i = blockIdx.x * blockDim.x + threadIdx.x;
    float bz = __expf(zp[0] * EPS_INV);
    if (i < NAUG) {
        K0[(size_t)MDIM * NAUG + i] = bz;   // bottom row (incl. corner)
        v[i] = 1.0f;
    }
    if (i < MDIM) {
        K0[(size_t)i * NAUG + MDIM] = bz;   // right column
    }
}

// u[i] = 1 / sum_j K0[i][j] * v[j]   (one block per row, tree reduction)
__global__ __launch_bounds__(256) void sinkhorn_row(
    const float* __restrict__ K0, const float* __restrict__ v,
    float* __restrict__ u)
{
    __shared__ float red[256];
    const int i = blockIdx.x;
    const float* row = K0 + (size_t)i * NAUG;
    float s = 0.0f;
    for (int j = threadIdx.x; j < NAUG; j += 256)
        s += row[j] * v[j];
    red[threadIdx.x] = s;
    __syncthreads();
    for (int off = 128; off > 0; off >>= 1) {
        if (threadIdx.x < off) red[threadIdx.x] += red[threadIdx.x + off];
        __syncthreads();
    }
    if (threadIdx.x == 0) u[i] = 1.0f / red[0];
}

// v[j] = 1 / sum_i K0[i][j] * u[i]   (thread-per-column, rows split 8 ways;
// column-major traversal is fully coalesced across threadIdx.x)
__global__ __launch_bounds__(512) void sinkhorn_col(
    const float* __restrict__ K0, const float* __restrict__ u,
    float* __restrict__ v)
{
    __shared__ float red[8][64];
    const int j = blockIdx.x * 64 + threadIdx.x;
    float s = 0.0f;
    if (j < NAUG) {
        for (int i = threadIdx.y; i < NAUG; i += 8)
            s += K0[(size_t)i * NAUG + j] * u[i];
    }
    red[threadIdx.y][threadIdx.x] = s;
    __syncthreads();
    if (threadIdx.y == 0 && j < NAUG) {
        float t = 0.0f;
        #pragma unroll
        for (int y = 0; y < 8; ++y) t += red[y][threadIdx.x];
        v[j] = 1.0f / t;
    }
}

// P_aug = diag(u) * K0 * diag(v) (in place); P = P_aug[:2048,:2048]
__global__ __launch_bounds__(256) void sinkhorn_finalize(
    float* __restrict__ Paug, float* __restrict__ P,
    const float* __restrict__ u, const float* __restrict__ v)
{
    const int j = blockIdx.x * blockDim.x + threadIdx.x;
    const int i = blockIdx.y;
    if (j < NAUG) {
        size_t idx = (size_t)i * NAUG + j;
        float val = Paug[idx] * u[i] * v[j];
        Paug[idx] = val;
        if (i < MDIM && j < MDIM) P[(size_t)i * MDIM + j] = val;
    }
}

extern "C" void kernel_launch(void* const* d_in, const int* in_sizes, int n_in,
                              void* d_out, int out_size, void* d_ws, size_t ws_size,
                              hipStream_t stream)
{
    const float* Q  = (const float*)d_in[0];   // (2048, 512) fp32
    const float* R  = (const float*)d_in[1];   // (2048, 512) fp32
    const float* zp = (const float*)d_in[2];   // scalar z

    float* P    = (float*)d_out;                 // 2048*2048
    float* Paug = P + (size_t)MDIM * MDIM;       // 2049*2049, doubles as K0
    float* u    = (float*)d_ws;                  // 2049 floats
    float* v    = u + 2304;                      // 2049 floats (padded offset)

    // K0 = exp(score/eps) into P_aug region
    sinkhorn_gemm_exp<<<dim3(32, 16), dim3(256), 0, stream>>>(Q, R, Paug);
    // borders = exp(z/eps), v = 1
    sinkhorn_init<<<dim3(9), dim3(256), 0, stream>>>(Paug, v, zp);

    // 100 Sinkhorn iterations via scaling vectors: u = 1/(K0 v); v = 1/(K0^T u)
    for (int it = 0; it < 100; ++it) {
        sinkhorn_row<<<dim3(NAUG), dim3(256), 0, stream>>>(Paug, v, u);
        sinkhorn_col<<<dim3(33), dim3(64, 8), 0, stream>>>(Paug, u, v);
    }

    // P_aug = diag(u) K0 diag(v); P = interior
    sinkhorn_finalize<<<dim3(9, NAUG), dim3(256), 0, stream>>>(Paug, P, u, v);
}